// STGCN_Layer_45938970198099
// MI455X (gfx1250) — compile-verified
//
#include <hip/hip_runtime.h>
#include <hip/hip_bf16.h>

typedef float v2f __attribute__((ext_vector_type(2)));
typedef float v8f __attribute__((ext_vector_type(8)));

#define N_BATCH 4
#define C_IN    64
#define T_DIM   64
#define V_DIM   25
#define OC      128
#define KS      3
#define M_ROWS  (N_BATCH * T_DIM * V_DIM)   // 6400
#define NTV_PER_N (T_DIM * V_DIM)           // 1600

__device__ __forceinline__ v8f wmma4(v2f a, v2f b, v8f c) {
    // D = A(16x4,f32) x B(4x16,f32) + C(16x16,f32)
    return __builtin_amdgcn_wmma_f32_16x16x4_f32(false, a, false, b, (short)0, c, false, false);
}

// ---------------------------------------------------------------------------
// x (n,c,t,v) -> xT (n,t,v,c) row-major feature vectors
__global__ void transpose_kernel(const float* __restrict__ x, float* __restrict__ xT) {
    int idx = blockIdx.x * 256 + threadIdx.x;
    if (idx >= M_ROWS * C_IN) return;
    int c = idx & 63;
    int m = idx >> 6;
    int n = m / NTV_PER_N;
    int rem = m - n * NTV_PER_N;                       // t*25+v
    xT[idx] = x[(size_t)n * (C_IN * NTV_PER_N) + (size_t)c * NTV_PER_N + rem];
}

// ---------------------------------------------------------------------------
// Fold w1x into gcn:  Wk[(k*OC+o), ci] = sum_c w1x[o,c]*gcn_w[k*OC+c, ci]
//                     bk[k*OC+o] = sum_c w1x[o,c]*gcn_b[k*OC+c] + b1[o]
__global__ void fold_kernel(const float* __restrict__ w1x, const float* __restrict__ gcn_w,
                            const float* __restrict__ gcn_b, const float* __restrict__ b1,
                            float* __restrict__ Wk, float* __restrict__ bk) {
    int b = blockIdx.x;            // k*128+o, 0..383
    int k = b >> 7, o = b & 127;
    int ci = threadIdx.x;          // 0..63
    __shared__ float sw[OC];
    for (int c = ci; c < OC; c += 64) sw[c] = w1x[o * OC + c];
    __syncthreads();
    float acc = 0.f;
    for (int c = 0; c < OC; ++c)
        acc += sw[c] * gcn_w[(size_t)(k * OC + c) * C_IN + ci];
    Wk[(size_t)b * C_IN + ci] = acc;
    if (ci == 0) {
        float bb = 0.f;
        for (int c = 0; c < OC; ++c) bb += sw[c] * gcn_b[k * OC + c];
        bk[b] = bb + b1[o];
    }
}

// S[v] = sum_{k,w} A[k,v,w]
__global__ void svec_kernel(const float* __restrict__ Amat, float* __restrict__ S) {
    int v = threadIdx.x;
    if (v < V_DIM) {
        float s = 0.f;
        for (int k = 0; k < KS; ++k)
            for (int w = 0; w < V_DIM; ++w)
                s += Amat[k * V_DIM * V_DIM + v * V_DIM + w];
        S[v] = s;
    }
}

// tcn_w (o,c,dt,1) -> twp[dt][o][c]
__global__ void pack_tcn_kernel(const float* __restrict__ tw, float* __restrict__ twp) {
    int idx = blockIdx.x * 256 + threadIdx.x;
    if (idx >= 9 * OC * OC) return;
    int c  = idx & 127;
    int o  = (idx >> 7) & 127;
    int dt = idx >> 14;
    twp[idx] = tw[((size_t)(o * OC + c)) * 9 + dt];
}

// ---------------------------------------------------------------------------
// Generic WMMA f32 GEMM:  Out[m, nn] = sum_k A[m,k]*W[nn,k] + bias
// bias = biasCol[nn] (optional)  +  biasCol2[nn]*svec[m % 25] (optional)
// grid.x = M/16, grid.y = Ntot/128, block = 256 (8 waves, one 16x16 tile each)
__global__ void gemm_wmma(const float* __restrict__ A, const float* __restrict__ W,
                          float* __restrict__ Out, int M, int Ntot, int K,
                          const float* __restrict__ biasCol,
                          const float* __restrict__ biasCol2,
                          const float* __restrict__ svec) {
    int lane = threadIdx.x & 31;
    int wave = threadIdx.x >> 5;
    int hi   = lane >> 4;          // 0/1
    int l15  = lane & 15;
    int m0 = blockIdx.x * 16;
    int n0 = (blockIdx.y * 8 + wave) * 16;
    if (n0 >= Ntot) return;        // wave-uniform

    int col = n0 + l15;
    const float* arow = A + (size_t)(m0 + l15) * K + 2 * hi;
    const float* wrow = W + (size_t)col * K + 2 * hi;

    float bc  = biasCol  ? biasCol[col]  : 0.f;
    float bc2 = biasCol2 ? biasCol2[col] : 0.f;
    v8f acc;
#pragma unroll
    for (int i = 0; i < 8; ++i) {
        float b = bc;
        if (biasCol2) {
            int row = m0 + 8 * hi + i;
            b += bc2 * svec[row % V_DIM];
        }
        acc[i] = b;
    }

#pragma unroll 16
    for (int k = 0; k < K; k += 4) {
        v2f a = *(const v2f*)(arow + k);
        v2f b = *(const v2f*)(wrow + k);
        acc = wmma4(a, b, acc);
    }

#pragma unroll
    for (int i = 0; i < 8; ++i) {
        int row = m0 + 8 * hi + i;
        Out[(size_t)row * Ntot + col] = acc[i];
    }
}

// ---------------------------------------------------------------------------
// Temporal conv (9,1) pad 4 as 9 shifted accumulated GEMMs, K=128 each.
// Q rows are (n,t,v); tap dt reads row m + (dt-4)*25, masked when t+dt-4 OOB.
// grid.x = M/16, block = 256 (8 waves cover N=128)
__global__ void gemm_tcn_wmma(const float* __restrict__ Q, const float* __restrict__ Wp,
                              float* __restrict__ Out) {
    int lane = threadIdx.x & 31;
    int wave = threadIdx.x >> 5;
    int hi = lane >> 4, l15 = lane & 15;
    int m0 = blockIdx.x * 16;
    int n0 = wave * 16;
    int rowA = m0 + l15;
    int t = (rowA % NTV_PER_N) / V_DIM;
    int col = n0 + l15;

    v8f acc;
#pragma unroll
    for (int i = 0; i < 8; ++i) acc[i] = 0.f;

    for (int dt = 0; dt < 9; ++dt) {
        int tt = t + dt - 4;
        bool valid = (tt >= 0) && (tt < T_DIM);
        int rs = valid ? (rowA + (dt - 4) * V_DIM) : rowA;   // clamp to safe addr
        const float* arow = Q + (size_t)rs * OC + 2 * hi;
        const float* wrow = Wp + ((size_t)dt * OC + col) * OC + 2 * hi;
#pragma unroll 32
        for (int k = 0; k < OC; k += 4) {
            v2f a = *(const v2f*)(arow + k);
            a[0] = valid ? a[0] : 0.f;
            a[1] = valid ? a[1] : 0.f;
            v2f b = *(const v2f*)(wrow + k);
            acc = wmma4(a, b, acc);
        }
    }
#pragma unroll
    for (int i = 0; i < 8; ++i)
        Out[(size_t)(m0 + 8 * hi + i) * OC + col] = acc[i];
}

// ---------------------------------------------------------------------------
// Fused pb + relu + A-fold:
// hA[(n,t,v), o] = sum_{k,w} A[k,v,w] * relu( px[(n,t,w), k*OC+o] + pb[n,t,v,w,o] )
// pb[...,o] = sum_b binary[n,t,v,w,b]*w1b[o,b]
// One block per (n,t); 128 threads (o).
__global__ void ha_kernel(const float* __restrict__ px, const float* __restrict__ binary,
                          const float* __restrict__ w1b, const float* __restrict__ Amat,
                          float* __restrict__ hA) {
    __shared__ float spx[V_DIM * KS * OC];   // [w][k*128+o]  9600 f
    __shared__ float sbin[V_DIM * V_DIM * 4];// [v][w][b]     2500 f
    __shared__ float sA[KS * V_DIM * V_DIM]; // [k][v][w]     1875 f
    int nt = blockIdx.x;                     // 0..255
    int o  = threadIdx.x;                    // 0..127
    size_t base = (size_t)nt * V_DIM;        // row of (n,t,0)

    for (int i = o; i < V_DIM * KS * OC; i += 128) spx[i] = px[base * (KS * OC) + i];
    for (int i = o; i < V_DIM * V_DIM * 4; i += 128) sbin[i] = binary[(size_t)nt * (V_DIM * V_DIM * 4) + i];
    for (int i = o; i < KS * V_DIM * V_DIM; i += 128) sA[i] = Amat[i];
    __syncthreads();

    float wb0 = w1b[o * 4 + 0], wb1 = w1b[o * 4 + 1];
    float wb2 = w1b[o * 4 + 2], wb3 = w1b[o * 4 + 3];

    for (int v = 0; v < V_DIM; ++v) {
        float acc = 0.f;
        for (int w = 0; w < V_DIM; ++w) {
            const float* bb = &sbin[(v * V_DIM + w) * 4];
            float pb = bb[0] * wb0 + bb[1] * wb1 + bb[2] * wb2 + bb[3] * wb3;
            float aw0 = sA[0 * 625 + v * V_DIM + w];
            float aw1 = sA[1 * 625 + v * V_DIM + w];
            float aw2 = sA[2 * 625 + v * V_DIM + w];
            const float* pp = &spx[w * (KS * OC) + o];
            acc += aw0 * fmaxf(pp[0]       + pb, 0.f);
            acc += aw1 * fmaxf(pp[OC]      + pb, 0.f);
            acc += aw2 * fmaxf(pp[2 * OC]  + pb, 0.f);
        }
        hA[(base + v) * OC + o] = acc;
    }
}

// ---------------------------------------------------------------------------
// Training-mode BN stats per channel -> scale/shift
__global__ void stats_kernel(const float* __restrict__ X, const float* __restrict__ gamma,
                             const float* __restrict__ beta, float* __restrict__ scale,
                             float* __restrict__ shift, int M) {
    int c = blockIdx.x, tid = threadIdx.x;  // 128 blocks x 256 threads
    float s = 0.f, s2 = 0.f;
    for (int m = tid; m < M; m += 256) {
        float x = X[(size_t)m * OC + c];
        s += x; s2 += x * x;
    }
    __shared__ float ss[256], ss2[256];
    ss[tid] = s; ss2[tid] = s2;
    __syncthreads();
    for (int st = 128; st > 0; st >>= 1) {
        if (tid < st) { ss[tid] += ss[tid + st]; ss2[tid] += ss2[tid + st]; }
        __syncthreads();
    }
    if (tid == 0) {
        float mean = ss[0] / (float)M;
        float var  = ss2[0] / (float)M - mean * mean;
        float a = gamma[c] * rsqrtf(var + 1e-5f);
        scale[c] = a;
        shift[c] = beta[c] - mean * a;
    }
}

// q = relu(g*scale + shift)
__global__ void bn_relu_kernel(const float* __restrict__ g, const float* __restrict__ sc,
                               const float* __restrict__ sh, float* __restrict__ q) {
    int idx = blockIdx.x * 256 + threadIdx.x;
    if (idx >= M_ROWS * OC) return;
    int c = idx & 127;
    q[idx] = fmaxf(g[idx] * sc[c] + sh[c], 0.f);
}

// out(n,c,t,v) = relu( bn2(z) + resbn(res) )   [(m,c) row-major -> NCHW]
__global__ void combine_kernel(const float* __restrict__ z, const float* __restrict__ res,
                               const float* __restrict__ a2, const float* __restrict__ c2,
                               const float* __restrict__ ar, const float* __restrict__ cr,
                               float* __restrict__ out) {
    int idx = blockIdx.x * 256 + threadIdx.x;
    if (idx >= M_ROWS * OC) return;
    int c = idx & 127;
    int m = idx >> 7;
    int n = m / NTV_PER_N;
    int rem = m - n * NTV_PER_N;
    float val = z[idx] * a2[c] + c2[c] + res[idx] * ar[c] + cr[c];
    out[(size_t)n * (OC * NTV_PER_N) + (size_t)c * NTV_PER_N + rem] = fmaxf(val, 0.f);
}

// ---------------------------------------------------------------------------
extern "C" void kernel_launch(void* const* d_in, const int* in_sizes, int n_in,
                              void* d_out, int out_size, void* d_ws, size_t ws_size,
                              hipStream_t stream) {
    const float* x      = (const float*)d_in[0];
    const float* Amat   = (const float*)d_in[1];
    const float* binary = (const float*)d_in[2];
    const float* gcn_w  = (const float*)d_in[3];
    const float* gcn_b  = (const float*)d_in[4];
    const float* w1x    = (const float*)d_in[5];
    const float* w1b    = (const float*)d_in[6];
    const float* b1     = (const float*)d_in[7];
    const float* w2     = (const float*)d_in[8];
    const float* b2     = (const float*)d_in[9];
    const float* tcn_w  = (const float*)d_in[10];
    const float* bn1_g  = (const float*)d_in[12];
    const float* bn1_b  = (const float*)d_in[13];
    const float* bn2_g  = (const float*)d_in[14];
    const float* bn2_b  = (const float*)d_in[15];
    const float* res_w  = (const float*)d_in[16];
    const float* resbn_g= (const float*)d_in[18];
    const float* resbn_b= (const float*)d_in[19];
    float* out = (float*)d_out;

    float* ws = (float*)d_ws;
    float* xT  = ws;                         // 409600
    float* px  = xT  + 409600;               // 2457600
    float* hA  = px  + 2457600;              // 819200
    float* g   = hA  + 819200;               // 819200
    float* q   = g   + 819200;               // 819200
    float* z   = q   + 819200;               // 819200
    float* res = z   + 819200;               // 819200
    float* Wk  = res + 819200;               // 24576
    float* bk  = Wk  + 24576;                // 384
    float* S   = bk  + 384;                  // 32
    float* twp = S   + 32;                   // 147456
    float* a1  = twp + 147456;
    float* c1  = a1 + 128;
    float* a2  = c1 + 128;
    float* c2  = a2 + 128;
    float* ar  = c2 + 128;
    float* cr  = ar + 128;

    // Prep
    transpose_kernel<<<1600, 256, 0, stream>>>(x, xT);
    fold_kernel<<<KS * OC, 64, 0, stream>>>(w1x, gcn_w, gcn_b, b1, Wk, bk);
    svec_kernel<<<1, 32, 0, stream>>>(Amat, S);
    pack_tcn_kernel<<<576, 256, 0, stream>>>(tcn_w, twp);

    // px[(n,t,v), k*OC+o] = xT @ Wk^T + bk   (M=6400, N=384, K=64)
    gemm_wmma<<<dim3(M_ROWS / 16, 3), 256, 0, stream>>>(xT, Wk, px, M_ROWS, KS * OC, C_IN,
                                                        bk, nullptr, nullptr);
    // residual (pre-BN; res_b is a per-channel constant -> vanishes under BN)
    gemm_wmma<<<dim3(M_ROWS / 16, 1), 256, 0, stream>>>(xT, res_w, res, M_ROWS, OC, C_IN,
                                                        nullptr, nullptr, nullptr);
    // fused pb + relu + A-fold
    ha_kernel<<<N_BATCH * T_DIM, 128, 0, stream>>>(px, binary, w1b, Amat, hA);
    // g = hA @ w2^T + b2*S[v]   (M=6400, N=128, K=128)
    gemm_wmma<<<dim3(M_ROWS / 16, 1), 256, 0, stream>>>(hA, w2, g, M_ROWS, OC, OC,
                                                        nullptr, b2, S);
    // bn1 stats + relu apply
    stats_kernel<<<OC, 256, 0, stream>>>(g, bn1_g, bn1_b, a1, c1, M_ROWS);
    bn_relu_kernel<<<3200, 256, 0, stream>>>(g, a1, c1, q);
    // tcn temporal conv (tcn_b vanishes under BN)
    gemm_tcn_wmma<<<M_ROWS / 16, 256, 0, stream>>>(q, twp, z);
    // bn2 / resbn stats, final combine -> NCHW output
    stats_kernel<<<OC, 256, 0, stream>>>(z, bn2_g, bn2_b, a2, c2, M_ROWS);
    stats_kernel<<<OC, 256, 0, stream>>>(res, resbn_g, resbn_b, ar, cr, M_ROWS);
    combine_kernel<<<3200, 256, 0, stream>>>(z, res, a2, c2, ar, cr, out);
}